// GCNNetwork_21397527068858
// MI455X (gfx1250) — compile-verified
//
#include <hip/hip_runtime.h>
#include <hip/hip_bf16.h>

// ---------------------------------------------------------------------------
// 2-layer GCN for MI455X (gfx1250, wave32).
//   Layer: h = x@W ; agg = segsum(norm[s]*norm[d]*h[s] -> d) + norm^2*h ; relu(agg+b)
// GEMM uses V_WMMA_F32_16X16X4_F32 (fp32 matrix path, 16x16 tiles).
// Edge scatter: one wave per edge, float4 per lane, global_atomic_add_f32.
// ---------------------------------------------------------------------------

typedef float v2f __attribute__((ext_vector_type(2)));
typedef float v8f __attribute__((ext_vector_type(8)));

#define D 128

// ---------------- utility: zero a float buffer ----------------
__global__ void fill_zero_kernel(float* __restrict__ p, long n) {
    long i = (long)blockIdx.x * blockDim.x + threadIdx.x;
    if (i < n) p[i] = 0.0f;
}

// ---------------- degree accumulation ----------------
__global__ void degree_kernel(const int* __restrict__ dst, float* __restrict__ deg, int nEdges) {
    int e = blockIdx.x * blockDim.x + threadIdx.x;
    if (e < nEdges) atomicAdd(&deg[dst[e]], 1.0f);
}

// ---------------- norm = rsqrt(deg + 1) in place ----------------
__global__ void norm_kernel(float* __restrict__ deg_norm, int nNodes) {
    int i = blockIdx.x * blockDim.x + threadIdx.x;
    if (i < nNodes) deg_norm[i] = rsqrtf(deg_norm[i] + 1.0f);
}

// ---------------- H[N,128] = X[N,128] @ W[128,128] via f32 WMMA ----------------
// Block = 256 threads = 8 waves. blockIdx.x selects a 16-row strip; wave w
// computes the 16x16 output tile at columns [16w, 16w+16). K advances by 4.
// nRows must be a multiple of 16 (100000 = 6250*16) so EXEC is all-1s.
__global__ void __launch_bounds__(256)
gemm_wmma_f32_kernel(const float* __restrict__ X, const float* __restrict__ W,
                     float* __restrict__ H) {
    const int wave = threadIdx.x >> 5;          // 0..7 -> column tile
    const int lane = threadIdx.x & 31;
    const int m    = lane & 15;                 // row/col within 16
    const int kOff = (lane >> 4) << 1;          // 0 (lanes 0-15) or 2 (lanes 16-31)
    const long row0 = (long)blockIdx.x * 16;
    const int  col0 = wave * 16;

    const float* __restrict__ xrow = X + (row0 + m) * D;   // A: row m of this strip

    v8f acc = {};
    #pragma unroll 8
    for (int k0 = 0; k0 < D; k0 += 4) {
        v2f a, b;
        // A 16x4 (MxK): VGPR0 = K {0|2}, VGPR1 = K {1|3} per lane half (ISA 7.12.2)
        a.x = xrow[k0 + kOff];
        a.y = xrow[k0 + kOff + 1];
        // B 4x16 (KxN): N striped across lanes, K split across lane halves
        b.x = W[(long)(k0 + kOff)     * D + col0 + m];
        b.y = W[(long)(k0 + kOff + 1) * D + col0 + m];
        acc = __builtin_amdgcn_wmma_f32_16x16x4_f32(
            /*neg_a=*/false, a, /*neg_b=*/false, b,
            /*c_mod=*/(short)0, acc, /*reuse_a=*/false, /*reuse_b=*/false);
    }

    // C/D 16x16 f32: VGPR r -> row r (lanes 0-15) or row r+8 (lanes 16-31)
    const long rbase = row0 + ((lane >> 4) << 3);
    #pragma unroll
    for (int r = 0; r < 8; ++r) {
        H[(rbase + r) * D + col0 + m] = acc[r];
    }
}

// ---------------- edge scatter: agg[dst] += norm[s]*norm[d] * h[src] ----------------
// One wave per edge; each lane handles 4 consecutive floats (128/32).
__global__ void __launch_bounds__(256)
edge_aggregate_kernel(const float* __restrict__ H, const int* __restrict__ src,
                      const int* __restrict__ dst, const float* __restrict__ norm,
                      float* __restrict__ agg, int nEdges) {
    const int e = blockIdx.x * (blockDim.x >> 5) + (threadIdx.x >> 5);
    if (e >= nEdges) return;
    const int lane = threadIdx.x & 31;
    const int s = src[e];
    const int d = dst[e];
    const float coef = norm[s] * norm[d];

    const float4 v = ((const float4*)(H + (long)s * D))[lane];
    float* __restrict__ ad = agg + (long)d * D + lane * 4;
    atomicAdd(ad + 0, v.x * coef);
    atomicAdd(ad + 1, v.y * coef);
    atomicAdd(ad + 2, v.z * coef);
    atomicAdd(ad + 3, v.w * coef);
}

// ---------------- epilogue: out = relu(agg + norm^2 * h + b) ----------------
__global__ void combine_kernel(const float* __restrict__ agg, const float* __restrict__ H,
                               const float* __restrict__ norm, const float* __restrict__ bias,
                               float* __restrict__ out, long total) {
    long i = (long)blockIdx.x * blockDim.x + threadIdx.x;
    if (i >= total) return;
    const long node = i >> 7;          // / 128
    const int  col  = (int)(i & 127);
    const float nn = norm[node];
    float v = agg[i] + H[i] * (nn * nn) + bias[col];
    out[i] = v > 0.0f ? v : 0.0f;
}

// ---------------------------------------------------------------------------
extern "C" void kernel_launch(void* const* d_in, const int* in_sizes, int n_in,
                              void* d_out, int out_size, void* d_ws, size_t ws_size,
                              hipStream_t stream) {
    const float* x    = (const float*)d_in[0];
    const int*   ei   = (const int*)d_in[1];
    const float* W1   = (const float*)d_in[2];
    const float* b1   = (const float*)d_in[3];
    const float* W2   = (const float*)d_in[4];
    const float* b2   = (const float*)d_in[5];

    const int nNodes = in_sizes[0] / D;      // 100000
    const int nEdges = in_sizes[1] / 2;      // 1600000
    const int* src = ei;
    const int* dst = ei + nEdges;

    const long ND = (long)nNodes * D;

    // Workspace: norm[N] | h[N*D] | agg[N*D]   (~103 MB)
    float* norm = (float*)d_ws;
    float* h    = norm + nNodes;
    float* agg  = h + ND;

    float* y = (float*)d_out;                // layer-1 activation lives in d_out

    const int T = 256;
    const int gridNodes   = (nNodes + T - 1) / T;
    const int gridEdges   = (nEdges + T - 1) / T;
    const int gridEdgeWav = (nEdges + (T / 32) - 1) / (T / 32);
    const int gridND      = (int)((ND + T - 1) / T);
    const int gridGemm    = nNodes / 16;     // 100000 % 16 == 0

    // --- degrees -> norm ---
    fill_zero_kernel<<<gridNodes, T, 0, stream>>>(norm, nNodes);
    degree_kernel<<<gridEdges, T, 0, stream>>>(dst, norm, nEdges);
    norm_kernel<<<gridNodes, T, 0, stream>>>(norm, nNodes);

    // --- layer 1 ---
    gemm_wmma_f32_kernel<<<gridGemm, T, 0, stream>>>(x, W1, h);
    fill_zero_kernel<<<gridND, T, 0, stream>>>(agg, ND);
    edge_aggregate_kernel<<<gridEdgeWav, T, 0, stream>>>(h, src, dst, norm, agg, nEdges);
    combine_kernel<<<gridND, T, 0, stream>>>(agg, h, norm, b1, y, ND);

    // --- layer 2 ---
    gemm_wmma_f32_kernel<<<gridGemm, T, 0, stream>>>(y, W2, h);
    fill_zero_kernel<<<gridND, T, 0, stream>>>(agg, ND);
    edge_aggregate_kernel<<<gridEdgeWav, T, 0, stream>>>(h, src, dst, norm, agg, nEdges);
    combine_kernel<<<gridND, T, 0, stream>>>(agg, h, norm, b2, y, ND);
}